// GCNModel_31662498906603
// MI455X (gfx1250) — compile-verified
//
#include <hip/hip_runtime.h>
#include <math.h>

typedef __attribute__((ext_vector_type(2))) float v2f;
typedef __attribute__((ext_vector_type(8))) float v8f;

#define FDIM 64   // F_IN == HIDDEN == 64
#define NCLS 40
#define NPAD 48

// ---------------- utility kernels ----------------

__global__ void zero_f32(float* __restrict__ p, long long n) {
  long long i = (long long)blockIdx.x * blockDim.x + threadIdx.x;
  long long stride = (long long)gridDim.x * blockDim.x;
  for (; i < n; i += stride) p[i] = 0.0f;
}

__global__ void pad_w2(const float* __restrict__ W2, float* __restrict__ w2p) {
  int i = blockIdx.x * blockDim.x + threadIdx.x; // 64*48 elements
  if (i >= FDIM * NPAD) return;
  int k = i / NPAD, c = i % NPAD;
  w2p[i] = (c < NCLS) ? W2[k * NCLS + c] : 0.0f;
}

__global__ void deg_count(const long long* __restrict__ dst, float* __restrict__ deg, int E) {
  int e = blockIdx.x * blockDim.x + threadIdx.x;
  if (e < E) atomicAdd(&deg[dst[e]], 1.0f);
}

__global__ void deg_to_dinv(float* __restrict__ deg, int N) {
  int n = blockIdx.x * blockDim.x + threadIdx.x;
  if (n < N) deg[n] = rsqrtf(deg[n] + 1.0f); // +1 self-loop
}

// ---------------- fp32 WMMA GEMM: C[M x LDC] = A[M x 64] * B[64 x LDB] ----------------
// One wave per 16-row stripe, NTILES 16-col tiles, K-loop of v_wmma_f32_16x16x4_f32.
template <int NTILES, int LDB, int LDC, bool MASKC>
__global__ void gemm_wmma_f32(const float* __restrict__ A, const float* __restrict__ B,
                              float* __restrict__ C, int M) {
  int wave = (int)(((long long)blockIdx.x * blockDim.x + threadIdx.x) >> 5);
  int rowBase = wave * 16;
  if (rowBase >= M) return;
  int lane = threadIdx.x & 31;
  int half = lane >> 4;   // 0: K = k0,k0+1 ; 1: K = k0+2,k0+3  (A & B layouts)
  int r    = lane & 15;

  const float* arow = A + (long long)(rowBase + r) * FDIM;

  v8f acc[NTILES] = {};

  for (int k0 = 0; k0 < FDIM; k0 += 4) {
    int ka = k0 + 2 * half;
    v2f a;
    a.x = arow[ka];
    a.y = arow[ka + 1];
#pragma unroll
    for (int t = 0; t < NTILES; ++t) {
      v2f b;
      b.x = B[ka * LDB + t * 16 + r];
      b.y = B[(ka + 1) * LDB + t * 16 + r];
      acc[t] = __builtin_amdgcn_wmma_f32_16x16x4_f32(
          /*neg_a=*/false, a, /*neg_b=*/false, b,
          /*c_mod=*/(short)0, acc[t], /*reuse_a=*/false, /*reuse_b=*/false);
    }
  }

#pragma unroll
  for (int t = 0; t < NTILES; ++t) {
#pragma unroll
    for (int i = 0; i < 8; ++i) {
      int row = rowBase + i + 8 * half; // D: vgpr i -> M=i (lanes 0-15), M=i+8 (lanes 16-31)
      int col = t * 16 + r;
      if (!MASKC || col < LDC) C[(long long)row * LDC + col] = acc[t][i];
    }
  }
}

// ---------------- edge scatter: one wave32 per edge ----------------

__global__ void scatter64(const long long* __restrict__ src, const long long* __restrict__ dst,
                          const float* __restrict__ dinv, const float* __restrict__ h,
                          float* __restrict__ agg, int E) {
  int e = (int)(((long long)blockIdx.x * blockDim.x + threadIdx.x) >> 5);
  if (e >= E) return;
  int lane = threadIdx.x & 31;
  long long s = src[e], d = dst[e];
  float norm = dinv[s] * dinv[d];
  const float* hs = h + s * 64;
  float* ad = agg + d * 64;
  atomicAdd(&ad[lane], hs[lane] * norm);
  atomicAdd(&ad[lane + 32], hs[lane + 32] * norm);
}

__global__ void scatter40(const long long* __restrict__ src, const long long* __restrict__ dst,
                          const float* __restrict__ dinv, const float* __restrict__ t2,
                          float* __restrict__ agg, int E) {
  int e = (int)(((long long)blockIdx.x * blockDim.x + threadIdx.x) >> 5);
  if (e >= E) return;
  int lane = threadIdx.x & 31;
  long long s = src[e], d = dst[e];
  float norm = dinv[s] * dinv[d];
  const float* ts = t2 + s * NCLS;
  float* ad = agg + d * NCLS;
  atomicAdd(&ad[lane], ts[lane] * norm);
  if (lane < 8) atomicAdd(&ad[lane + 32], ts[lane + 32] * norm);
}

// ---------------- self-loop + bias + ReLU (layer 1 epilogue) ----------------

__global__ void self_bias_relu(const float* __restrict__ h, const float* __restrict__ dinv,
                               const float* __restrict__ b1, float* __restrict__ agg,
                               long long total) {
  long long i = (long long)blockIdx.x * blockDim.x + threadIdx.x;
  if (i >= total) return;
  long long n = i >> 6;
  int f = (int)(i & 63);
  float di = dinv[n];
  float v = agg[i] + h[i] * (di * di) + b1[f];
  agg[i] = v > 0.0f ? v : 0.0f;
}

// ---------------- layer-2 epilogue + log_softmax: one wave32 per node ----------------

__global__ void final_lsm(const float* __restrict__ t2, const float* __restrict__ dinv,
                          const float* __restrict__ b2, float* __restrict__ out, int N) {
  int n = (int)(((long long)blockIdx.x * blockDim.x + threadIdx.x) >> 5);
  if (n >= N) return;
  int lane = threadIdx.x & 31;
  float dd = dinv[n] * dinv[n];
  long long base = (long long)n * NCLS;
  float v0 = out[base + lane] + t2[base + lane] * dd + b2[lane];
  float v1 = -INFINITY;
  if (lane < 8) v1 = out[base + 32 + lane] + t2[base + 32 + lane] * dd + b2[32 + lane];

  float m = fmaxf(v0, v1);
  for (int off = 16; off > 0; off >>= 1) m = fmaxf(m, __shfl_xor(m, off, 32));
  float s = expf(v0 - m) + (lane < 8 ? expf(v1 - m) : 0.0f);
  for (int off = 16; off > 0; off >>= 1) s += __shfl_xor(s, off, 32);
  float lse = m + logf(s);

  out[base + lane] = v0 - lse;
  if (lane < 8) out[base + 32 + lane] = v1 - lse;
}

// ---------------- launcher ----------------

extern "C" void kernel_launch(void* const* d_in, const int* in_sizes, int n_in,
                              void* d_out, int out_size, void* d_ws, size_t ws_size,
                              hipStream_t stream) {
  const float* x      = (const float*)d_in[0];
  const long long* ei = (const long long*)d_in[1];
  const float* W1     = (const float*)d_in[2];
  const float* b1     = (const float*)d_in[3];
  const float* W2     = (const float*)d_in[4];
  const float* b2     = (const float*)d_in[5];

  const int N = in_sizes[0] / FDIM;
  const int E = in_sizes[1] / 2;
  const long long* src = ei;
  const long long* dst = ei + E;

  float* ws   = (float*)d_ws;
  float* deg  = ws;                          // N (degree -> dinv in place)
  float* h    = deg + N;                     // N*64 : x @ W1
  float* agg1 = h + (size_t)N * 64;          // N*64 : layer-1 aggregate -> relu'd h1
  float* t2   = agg1 + (size_t)N * 64;       // N*40 : h1 @ W2
  float* w2p  = t2 + (size_t)N * NCLS;       // 64*48 : zero-padded W2
  float* out  = (float*)d_out;               // N*40 : layer-2 aggregate, then log_softmax

  // init (deterministic every call; d_out/d_ws may be poisoned)
  zero_f32<<<1024, 256, 0, stream>>>(deg, N);
  zero_f32<<<4096, 256, 0, stream>>>(agg1, (long long)N * 64);
  zero_f32<<<4096, 256, 0, stream>>>(out, (long long)N * NCLS);
  pad_w2<<<(FDIM * NPAD + 255) / 256, 256, 0, stream>>>(W2, w2p);

  // degrees (self-loop added in rsqrt pass)
  deg_count<<<(E + 255) / 256, 256, 0, stream>>>(dst, deg, E);
  deg_to_dinv<<<(N + 255) / 256, 256, 0, stream>>>(deg, N);

  const int waves = (N + 15) / 16;          // one wave per 16 output rows
  const int gemmBlocks = (waves + 7) / 8;   // 8 waves (256 threads) per block
  const int edgeBlocks = (E + 7) / 8;       // 8 edges (one per wave) per block

  // layer 1: h = x @ W1 ; aggregate ; + self-loop + b1 ; relu
  gemm_wmma_f32<4, 64, 64, false><<<gemmBlocks, 256, 0, stream>>>(x, W1, h, N);
  scatter64<<<edgeBlocks, 256, 0, stream>>>(src, dst, deg, h, agg1, E);
  self_bias_relu<<<(int)(((long long)N * 64 + 255) / 256), 256, 0, stream>>>(
      h, deg, b1, agg1, (long long)N * 64);

  // layer 2: t2 = h1 @ W2 (padded to 48 cols, masked store) ; aggregate ; epilogue + log_softmax
  gemm_wmma_f32<3, NPAD, NCLS, true><<<gemmBlocks, 256, 0, stream>>>(agg1, w2p, t2, N);
  scatter40<<<edgeBlocks, 256, 0, stream>>>(src, dst, deg, t2, out, E);
  final_lsm<<<(N + 7) / 8, 256, 0, stream>>>(t2, deg, b2, out, N);
}